// MeshPoolBlock_90486370993027
// MI455X (gfx1250) — compile-verified
//
#include <hip/hip_runtime.h>
#include <hip/hip_bf16.h>

typedef float v2f __attribute__((ext_vector_type(2)));
typedef float v8f __attribute__((ext_vector_type(8)));

// ---------------------------------------------------------------------------
// Phase 0: initialize the per-column argmin keys to all-ones (= +inf, idx max).
// ---------------------------------------------------------------------------
__global__ void nn_init_keys(unsigned long long* __restrict__ keys, int M) {
    int i = blockIdx.x * blockDim.x + threadIdx.x;
    if (i < M) keys[i] = ~0ULL;
}

// ---------------------------------------------------------------------------
// Phase 1: WMMA-based nearest-neighbor argmin.
//
// One wave (32 threads) owns a 16-column tile of sub_vertices and scans a
// chunk of vertex tiles (16 rows each). K=4 trick:
//   A row  i = ( vx, vy, vz, |v|^2 )        (16x4 f32)
//   B col  j = (-2sx,-2sy,-2sz, 1 )         (4x16 f32)
//   D[i][j]  = |v_i|^2 - 2 v_i . s_j  =  d2(i,j) - |s_j|^2
// Argmin over i of D equals argmin of the true squared distance.
//
// Tail handling: rows with n >= N get |v|^2 := +inf, so the WMMA emits +inf
// for the whole row and the strict-< argmin never selects it. No per-row
// bounds checks needed in the reduce loop.
//
// A layout (32-bit 16x4): lanes 0-15 hold K=0 (V0), K=1 (V1) for M=lane;
//                         lanes 16-31 hold K=2 (V0), K=3 (V1) for M=lane-16.
// B layout (4x16):        V0: lanes 0-15 K=0, lanes 16-31 K=2
//                         V1: lanes 0-15 K=1, lanes 16-31 K=3
// D layout: VGPR r: lanes 0-15 -> row r, col lane; lanes 16-31 -> row r+8.
// ---------------------------------------------------------------------------
__global__ __launch_bounds__(32)
void nn_argmin_kernel(const float* __restrict__ vert,
                      const float* __restrict__ sub,
                      unsigned long long* __restrict__ keys,
                      int N, int M, int tilesPerChunk, int nTilesTotal) {
    const int lane = threadIdx.x & 31;
    const int l16  = lane & 15;
    const int half = lane >> 4;

    const int m0   = blockIdx.x * 16;
    const int mIdx = m0 + l16;
    const int mC   = (mIdx < M) ? mIdx : (M - 1);

    const float sx = sub[3 * mC + 0];
    const float sy = sub[3 * mC + 1];
    const float sz = sub[3 * mC + 2];

    v2f b;
    b.x = half ? (-2.0f * sz) : (-2.0f * sx);
    b.y = half ? 1.0f         : (-2.0f * sy);

    float bestVal = __builtin_inff();
    int   bestN   = 0;

    const int tStart = blockIdx.y * tilesPerChunk;
    int tEnd = tStart + tilesPerChunk;
    if (tEnd > nTilesTotal) tEnd = nTilesTotal;
    if (tStart >= tEnd) tEnd = tStart;   // empty chunk -> skip loop

    // --- software-pipelined vertex tile load -------------------------------
    float vx = 0.f, vy = 0.f, vz = 0.f;
    bool  inR = false;
    if (tStart < tEnd) {
        const int nIdx = tStart * 16 + l16;
        const int nC   = (nIdx < N) ? nIdx : (N - 1);
        vx = vert[3 * nC + 0];
        vy = vert[3 * nC + 1];
        vz = vert[3 * nC + 2];
        inR = (nIdx < N);
    }

    for (int t = tStart; t < tEnd; ++t) {
        const int n0 = t * 16;

        // Build A from the tile loaded for iteration t.
        const float vn2 = inR ? (vx * vx + vy * vy + vz * vz)
                              : __builtin_inff();   // poison OOB rows
        v2f a;
        a.x = half ? vz  : vx;
        a.y = half ? vn2 : vy;

        // Prefetch tile t+1 (overlaps the WMMA + reduce below).
        if (t + 1 < tEnd) {
            const int nIdx = (t + 1) * 16 + l16;
            const int nC   = (nIdx < N) ? nIdx : (N - 1);
            vx = vert[3 * nC + 0];
            vy = vert[3 * nC + 1];
            vz = vert[3 * nC + 2];
            inR = (nIdx < N);
        }

        v8f c = {0.f, 0.f, 0.f, 0.f, 0.f, 0.f, 0.f, 0.f};
        // 8 args: (neg_a, A, neg_b, B, c_mod, C, reuse_a, reuse_b)
        c = __builtin_amdgcn_wmma_f32_16x16x4_f32(
                false, a, false, b, (short)0, c, false, false);

        #pragma unroll
        for (int r = 0; r < 8; ++r) {
            const int   n   = n0 + half * 8 + r;
            const float val = c[r];
            const bool take = val < bestVal;   // strict: keeps lowest n on ties
            bestN   = take ? n   : bestN;
            bestVal = take ? val : bestVal;
        }
    }

    // Merge the two half-wave results for each column (lane j <-> lane j+16).
    const float oVal = __shfl_xor(bestVal, 16, 32);
    const int   oN   = __shfl_xor(bestN,   16, 32);
    if (oVal < bestVal || (oVal == bestVal && oN < bestN)) {
        bestVal = oVal;
        bestN   = oN;
    }

    if (lane < 16 && mIdx < M) {
        // Map float to order-preserving unsigned bits (handles negatives).
        unsigned ub = __float_as_uint(bestVal);
        ub = (ub & 0x80000000u) ? ~ub : (ub | 0x80000000u);
        const unsigned long long key =
            ((unsigned long long)ub << 32) | (unsigned)bestN;
        atomicMin(&keys[mIdx], key);
    }
}

// ---------------------------------------------------------------------------
// Phase 2: gather feature rows: out[m, :] = X[argmin_n d2(n, m), :]
// ---------------------------------------------------------------------------
__global__ void nn_gather_kernel(const float* __restrict__ X,
                                 const unsigned long long* __restrict__ keys,
                                 float* __restrict__ out, int F) {
    const int m = blockIdx.x;
    const unsigned n = (unsigned)(keys[m] & 0xFFFFFFFFu);
    const float* srcRow = X   + (size_t)n * (size_t)F;
    float*       dstRow = out + (size_t)m * (size_t)F;
    if ((F & 3) == 0) {
        const float4* src4 = (const float4*)srcRow;
        float4*       dst4 = (float4*)dstRow;
        const int F4 = F >> 2;
        for (int t = threadIdx.x; t < F4; t += blockDim.x) dst4[t] = src4[t];
    } else {
        for (int t = threadIdx.x; t < F; t += blockDim.x) dstRow[t] = srcRow[t];
    }
}

// ---------------------------------------------------------------------------
// Launcher
// ---------------------------------------------------------------------------
extern "C" void kernel_launch(void* const* d_in, const int* in_sizes, int n_in,
                              void* d_out, int out_size, void* d_ws, size_t ws_size,
                              hipStream_t stream) {
    const float* vert = (const float*)d_in[0];   // [N, 3]
    const float* sub  = (const float*)d_in[1];   // [M, 3]
    const float* X    = (const float*)d_in[2];   // [N, F]
    float* out = (float*)d_out;                  // [M, F]

    const int N = in_sizes[0] / 3;
    const int M = in_sizes[1] / 3;
    const int F = in_sizes[2] / N;

    unsigned long long* keys = (unsigned long long*)d_ws;  // M * 8 bytes

    const int mTiles = (M + 15) / 16;
    const int nTiles = (N + 15) / 16;
    int chunks = 25;                       // split N for occupancy
    if (chunks > nTiles) chunks = nTiles;
    const int tilesPerChunk = (nTiles + chunks - 1) / chunks;

    nn_init_keys<<<(M + 255) / 256, 256, 0, stream>>>(keys, M);
    nn_argmin_kernel<<<dim3(mTiles, chunks), 32, 0, stream>>>(
        vert, sub, keys, N, M, tilesPerChunk, nTiles);
    nn_gather_kernel<<<M, 64, 0, stream>>>(X, keys, out, F);
}